// LSTMCell_61254823576021
// MI455X (gfx1250) — compile-verified
//
#include <hip/hip_runtime.h>

#define Bn   128
#define Ln   1024
#define Fn   128
#define Hn   64
#define NCn  16
#define G4   256              // 4*H gates
#define K1   65               // H+1 input length
#define K1P  68               // padded row length (16B-aligned, /4)
#define IN_DIM (Fn*Hn + Hn)   // 8256

typedef __attribute__((ext_vector_type(2))) float v2f;
typedef __attribute__((ext_vector_type(8))) float v8f;

__device__ __forceinline__ float sigmoidf_(float x) {
    return 1.0f / (1.0f + __expf(-x));
}

// ---------------------------------------------------------------------------
// Kernel 0 (one-time): repack W_gates (F,256,65) -> (F,256,68) zero-padded so
// hot-loop rows are 16B-aligned and fully coverable by global_load_b128.
// ---------------------------------------------------------------------------
__global__ __launch_bounds__(256) void repack_w(
    const float* __restrict__ Wg, float* __restrict__ Wp)
{
    const size_t total = (size_t)Fn * G4 * K1P;
    size_t idx = (size_t)blockIdx.x * 256 + threadIdx.x;
    if (idx < total) {
        const size_t r = idx / K1P;
        const int    k = (int)(idx % K1P);
        Wp[idx] = (k < K1) ? Wg[r * K1 + k] : 0.0f;
    }
}

// ---------------------------------------------------------------------------
// Kernel 1: event-wise LSTM recurrence. One workgroup (256 thr = 8 wave32)
// per sample; thread g owns gate g. h (32KB), c, gates, and the full event
// stream (async-copied global->LDS) live in LDS. Padded W tiles stream from
// L2 as b128 clauses, with the *next* step's tile prefetched each iteration.
// ---------------------------------------------------------------------------
__global__ __launch_bounds__(256) void evlstm_recur(
    const float* __restrict__ X, const int* __restrict__ lengths,
    const float* __restrict__ Wp, const float* __restrict__ bg,
    const float* __restrict__ wd, const float* __restrict__ bd,
    float* __restrict__ feat)
{
    __shared__ float h_s[Fn * Hn];    // 32 KB
    __shared__ float c_s[Hn];
    __shared__ float inp_s[K1P];      // [x, decay*h[mj], 0,0,0]
    __shared__ float gact_s[G4];      // activated gates
    __shared__ float ev_m[Ln];        // event stream (12 KB)
    __shared__ float ev_x[Ln];
    __shared__ float ev_d[Ln];
    __shared__ float wd_s[Fn], bd_s[Fn];

    const int b = blockIdx.x;
    const int g = threadIdx.x;
    const int len = lengths[b];
    const float* Xb = X + (size_t)b * 4 * Ln;   // X[b, ch, l]

    // ---- one-time preload of the event stream into LDS (ASYNCcnt path) ----
#if __has_builtin(__builtin_amdgcn_global_load_async_to_lds_b32)
    for (int i = g; i < Ln; i += 256) {
        __builtin_amdgcn_global_load_async_to_lds_b32(
            (int*)(Xb + 1 * Ln + i), (int*)(&ev_m[i]), 0, 0);
        __builtin_amdgcn_global_load_async_to_lds_b32(
            (int*)(Xb + 2 * Ln + i), (int*)(&ev_x[i]), 0, 0);
        __builtin_amdgcn_global_load_async_to_lds_b32(
            (int*)(Xb + 3 * Ln + i), (int*)(&ev_d[i]), 0, 0);
    }
#if __has_builtin(__builtin_amdgcn_s_wait_asynccnt)
    __builtin_amdgcn_s_wait_asynccnt(0);
#else
    asm volatile("s_wait_asynccnt 0" ::: "memory");
#endif
#else
    for (int i = g; i < Ln; i += 256) {
        ev_m[i] = Xb[1 * Ln + i];
        ev_x[i] = Xb[2 * Ln + i];
        ev_d[i] = Xb[3 * Ln + i];
    }
#endif

    for (int i = g; i < Fn * Hn; i += 256) h_s[i] = 0.0f;
    if (g < Hn) c_s[g] = 0.0f;
    if (g < Fn) { wd_s[g] = wd[g]; bd_s[g] = bd[g]; }
    if (g >= K1 && g < K1P) inp_s[g] = 0.0f;     // zero pad lanes
    __syncthreads();

    for (int t = 0; t < Ln; ++t) {
        const int   mj = (int)ev_m[t];
        const float xj = ev_x[t];
        const float dj = ev_d[t];
        const bool  valid = (t < len);

        // Prefetch next step's padded W tile (544 cachelines) + bias row.
        if (t + 1 < Ln) {
            const int mn = (int)ev_m[t + 1];
            const char* tb = (const char*)(Wp + (size_t)mn * G4 * K1P);
            for (int k = g; k < (G4 * K1P * 4) / 128; k += 256)
                __builtin_prefetch(tb + (size_t)k * 128, 0, 3);
            if (g < 8)
                __builtin_prefetch(bg + (size_t)mn * G4 + g * 32, 0, 3);
        }

        // Build input vector [x, decay * h[mj]]
        if (g < Hn) {
            const float dec = __expf(-fmaxf(0.0f, fmaf(wd_s[mj], dj, bd_s[mj])));
            inp_s[1 + g] = dec * h_s[mj * Hn + g];
        } else if (g == Hn) {
            inp_s[0] = xj;
        }
        __syncthreads();

        // gate_g = b_g + W_row(g) . inp   (17 b128 loads + 68 FMAs)
        const float4* W4 = (const float4*)(Wp + ((size_t)mj * G4 + g) * K1P);
        float acc = bg[(size_t)mj * G4 + g];
        #pragma unroll
        for (int k4 = 0; k4 < K1P / 4; ++k4) {
            const float4 w = W4[k4];
            acc = fmaf(w.x, inp_s[4 * k4 + 0], acc);
            acc = fmaf(w.y, inp_s[4 * k4 + 1], acc);
            acc = fmaf(w.z, inp_s[4 * k4 + 2], acc);
            acc = fmaf(w.w, inp_s[4 * k4 + 3], acc);
        }

        // order: [gi(0:64) gf(64:128) go(128:192) gc(192:256)]
        gact_s[g] = (g < 3 * Hn) ? sigmoidf_(acc) : tanhf(acc);
        __syncthreads();

        if (g < Hn && valid) {
            const float cn = fmaf(gact_s[Hn + g], c_s[g],
                                  gact_s[g] * gact_s[3 * Hn + g]);
            c_s[g] = cn;
            h_s[mj * Hn + g] = gact_s[2 * Hn + g] * tanhf(cn);
        }
        __syncthreads();
    }

    // feat[b] = concat(c (64), h (8192))
    float* fb = feat + (size_t)b * IN_DIM;
    if (g < Hn) fb[g] = c_s[g];
    for (int i = g; i < Fn * Hn; i += 256) fb[Hn + i] = h_s[i];
}

// ---------------------------------------------------------------------------
// Kernel 2: logits = feat(128x8256) @ W_out^T(8256x16) + b, then softmax.
// One wave per 16-row tile; V_WMMA_F32_16X16X4_F32 accumulate over K.
// ---------------------------------------------------------------------------
__global__ __launch_bounds__(32) void out_gemm_softmax(
    const float* __restrict__ feat, const float* __restrict__ Wout,
    const float* __restrict__ bout, float* __restrict__ out)
{
    __shared__ float lg[16][17];
    const int lane = threadIdx.x;
    const int n  = lane & 15;        // M for A loads, N for B loads
    const int hi = lane >> 4;        // selects K pair {0,1} vs {2,3}
    const int r0 = blockIdx.x * 16;

    const float* A0 = feat + (size_t)(r0 + n) * IN_DIM + 2 * hi;
    const float* B0 = Wout + (size_t)n * IN_DIM + 2 * hi;

    v8f c = {0.f, 0.f, 0.f, 0.f, 0.f, 0.f, 0.f, 0.f};
    for (int kc = 0; kc < IN_DIM; kc += 4) {     // 2064 WMMA steps
        v2f a, bm;
        a.x  = A0[kc];  a.y  = A0[kc + 1];
        bm.x = B0[kc];  bm.y = B0[kc + 1];
        c = __builtin_amdgcn_wmma_f32_16x16x4_f32(
                false, a, false, bm, (short)0, c, false, false);
    }

    #pragma unroll
    for (int v = 0; v < 8; ++v)
        lg[v + 8 * hi][n] = c[v] + bout[n];
    __syncthreads();

    if (lane < 16) {
        float mx = lg[lane][0];
        #pragma unroll
        for (int j = 1; j < 16; ++j) mx = fmaxf(mx, lg[lane][j]);
        float e[16], s = 0.f;
        #pragma unroll
        for (int j = 0; j < 16; ++j) { e[j] = __expf(lg[lane][j] - mx); s += e[j]; }
        const float inv = 1.0f / s;
        #pragma unroll
        for (int j = 0; j < 16; ++j)
            out[(size_t)(r0 + lane) * NCn + j] = e[j] * inv;
    }
}

// ---------------------------------------------------------------------------
extern "C" void kernel_launch(void* const* d_in, const int* in_sizes, int n_in,
                              void* d_out, int out_size, void* d_ws, size_t ws_size,
                              hipStream_t stream) {
    const float* X      = (const float*)d_in[0];
    const int*   lens   = (const int*)  d_in[1];
    const float* Wg     = (const float*)d_in[2];
    const float* bg     = (const float*)d_in[3];
    const float* wd     = (const float*)d_in[4];
    const float* bd     = (const float*)d_in[5];
    const float* Wout   = (const float*)d_in[6];
    const float* bout   = (const float*)d_in[7];

    // Workspace layout: [feat: 128*8256 f32 = 4.23 MB][Wpad: 128*256*68 f32 = 8.91 MB]
    float* feat = (float*)d_ws;
    float* Wpad = (float*)((char*)d_ws + (size_t)Bn * IN_DIM * sizeof(float));

    const size_t wtot = (size_t)Fn * G4 * K1P;
    repack_w<<<(int)((wtot + 255) / 256), 256, 0, stream>>>(Wg, Wpad);
    evlstm_recur<<<Bn, 256, 0, stream>>>(X, lens, Wpad, bg, wd, bd, feat);
    out_gemm_softmax<<<Bn / 16, 32, 0, stream>>>(feat, Wout, bout, (float*)d_out);
}